// MultiboxLoss_1219770712096
// MI455X (gfx1250) — compile-verified
//
#include <hip/hip_runtime.h>
#include <hip/hip_bf16.h>

#define BATCH 64
#define TGT   100
#define NA    8732
#define THRESH 0.5f

#define SEGS      7      // anchor segments per batch
#define TILES     5      // 256-anchor tiles per segment
#define SEGLEN    (TILES * 256)   // 1280; 7*1280 = 8960 >= 8732

typedef int v4i __attribute__((ext_vector_type(4)));
typedef __attribute__((address_space(1))) v4i* gptr_v4i;
typedef __attribute__((address_space(3))) v4i* lptr_v4i;

// ---- CDNA5 async global->LDS path (guarded; falls back to direct copy) ----
#ifndef USE_ASYNC_LDS
#if defined(__HIP_DEVICE_COMPILE__) && defined(__has_builtin)
#if __has_builtin(__builtin_amdgcn_global_load_async_to_lds_b128) && \
    __has_builtin(__builtin_amdgcn_s_wait_asynccnt)
#define USE_ASYNC_LDS 1
#endif
#endif
#endif
#ifndef USE_ASYNC_LDS
#define USE_ASYNC_LDS 0
#endif

__device__ __forceinline__ void cp16_g2lds(const float* g, float* l) {
#if USE_ASYNC_LDS
  float* gnc = const_cast<float*>(g);
  __builtin_amdgcn_global_load_async_to_lds_b128(
      (gptr_v4i)gnc, (lptr_v4i)l, /*ioffset=*/0, /*cpol=*/0);
#else
  *(float4*)l = *(const float4*)g;
#endif
}
__device__ __forceinline__ void wait_async0() {
#if USE_ASYNC_LDS
  __builtin_amdgcn_s_wait_asynccnt(0);
#endif
}
__device__ __forceinline__ void wait_async1() {
#if USE_ASYNC_LDS
  __builtin_amdgcn_s_wait_asynccnt(1);
#endif
}

// Stage targets[b] (500 floats) into LDS raw buffer via async DMA.
__device__ __forceinline__ void stage_targets(const float* tg, int b, int tid,
                                              float4* traw4) {
  if (tid < 125)
    cp16_g2lds(tg + (size_t)b * (TGT * 5) + tid * 4, (float*)&traw4[tid]);
  wait_async0();
  __syncthreads();
}

// ---- Kernel A: zero the per-(b,t) atomic argmax keys ----
__global__ void init_best_kernel(unsigned long long* __restrict__ wsbest) {
  int i = blockIdx.x * blockDim.x + threadIdx.x;
  if (i < BATCH * TGT) wsbest[i] = 0ull;
}

// ---- Kernel B: per-anchor argmax over targets + per-target argmax over
//      anchors (segment-local, finished with one global u64 atomic-max). ----
__global__ __launch_bounds__(256) void match_kernel(
    const float* __restrict__ tg, const float* __restrict__ an,
    float* __restrict__ wsov, int* __restrict__ wsix,
    unsigned long long* __restrict__ wsbest) {
  __shared__ float4 buf[2][256];           // double-buffered anchor tiles
  __shared__ float4 tb[TGT];
  __shared__ float  tarea[TGT];
  __shared__ unsigned long long lbest[TGT];
  __shared__ float4 traw4[125];

  const int tid  = threadIdx.x;
  const int seg  = blockIdx.x;
  const int b    = blockIdx.y;
  const int bA   = b * NA;
  const int base = seg * SEGLEN;

  stage_targets(tg, b, tid, traw4);
  const float* traw = (const float*)traw4;
  if (tid < TGT) {
    float x1 = traw[tid * 5 + 0], y1 = traw[tid * 5 + 1];
    float x2 = traw[tid * 5 + 2], y2 = traw[tid * 5 + 3];
    tb[tid]    = make_float4(x1, y1, x2, y2);
    tarea[tid] = (x2 - x1) * (y2 - y1);
    lbest[tid] = 0ull;
  }
  __syncthreads();

  // preload tile 0 of this segment
  {
    int g = base + tid; if (g >= NA) g = NA - 1;
    cp16_g2lds(an + (size_t)g * 4, (float*)&buf[0][tid]);
  }
  for (int k = 0; k < TILES; ++k) {
    const int cur = k & 1;
    if (k + 1 < TILES) {
      int g = base + (k + 1) * 256 + tid; if (g >= NA) g = NA - 1;
      cp16_g2lds(an + (size_t)g * 4, (float*)&buf[cur ^ 1][tid]);
      wait_async1();       // tile k complete (async loads retire in order)
    } else {
      wait_async0();
    }
    __syncthreads();

    // per-anchor max/argmax over targets (first-max tie-break = jnp.argmax)
    const int a = base + k * 256 + tid;
    if (a < NA) {
      float4 c = buf[cur][tid];
      float ax1 = c.x - 0.5f * c.z, ay1 = c.y - 0.5f * c.w;
      float ax2 = c.x + 0.5f * c.z, ay2 = c.y + 0.5f * c.w;
      float aarea = c.z * c.w;
      float bi = -1.0f; int bt = 0;
#pragma unroll 4
      for (int t = 0; t < TGT; ++t) {
        float4 m = tb[t];
        float iw = fmaxf(fminf(m.z, ax2) - fmaxf(m.x, ax1), 0.0f);
        float ih = fmaxf(fminf(m.w, ay2) - fmaxf(m.y, ay1), 0.0f);
        float inter = iw * ih;
        float iou = inter / (tarea[t] + aarea - inter);
        if (iou > bi) { bi = iou; bt = t; }
      }
      wsov[bA + a] = bi;
      wsix[bA + a] = bt;
    }

    // per-target max over this tile (wave32 shuffle reduction)
    const int wv = tid >> 5, lane = tid & 31;
    const int tilebase = base + k * 256;
    for (int t = wv; t < TGT; t += 8) {
      float4 m = tb[t];
      float  ab = tarea[t];
      unsigned long long key = 0ull;
      for (int al = lane; al < 256; al += 32) {
        int g = tilebase + al;
        if (g < NA) {
          float4 c = buf[cur][al];
          float ax1 = c.x - 0.5f * c.z, ay1 = c.y - 0.5f * c.w;
          float ax2 = c.x + 0.5f * c.z, ay2 = c.y + 0.5f * c.w;
          float iw = fmaxf(fminf(m.z, ax2) - fmaxf(m.x, ax1), 0.0f);
          float ih = fmaxf(fminf(m.w, ay2) - fmaxf(m.y, ay1), 0.0f);
          float inter = iw * ih;
          float iou = inter / (ab + c.z * c.w - inter);
          unsigned long long kk =
              ((unsigned long long)__float_as_uint(iou) << 32) |
              (unsigned int)(NA - 1 - g);      // smaller anchor index wins ties
          if (kk > key) key = kk;
        }
      }
#pragma unroll
      for (int off = 16; off > 0; off >>= 1) {
        unsigned long long other = __shfl_xor(key, off, 32);
        if (other > key) key = other;
      }
      if (lane == 0 && key > lbest[t]) lbest[t] = key;  // only wave wv writes t
    }
    __syncthreads();
  }

  // fold segment-local winners into the global per-(b,t) key
  if (tid < TGT && lbest[tid] != 0ull)
    atomicMax(&wsbest[b * TGT + tid], lbest[tid]);
}

// ---- Kernel C: apply per-target overrides (last-writer-wins via forward
//      scan) and emit loc/conf. ----
__global__ __launch_bounds__(256) void encode_kernel(
    const float* __restrict__ tg, const float* __restrict__ an,
    const float* __restrict__ wsov, const int* __restrict__ wsix,
    const unsigned long long* __restrict__ wsbest,
    float* __restrict__ loc, int* __restrict__ conf) {
  __shared__ float4 tb[TGT];
  __shared__ float  tlabel[TGT];
  __shared__ int    aidx[TGT];
  __shared__ float4 traw4[125];

  const int tid  = threadIdx.x;
  const int seg  = blockIdx.x;
  const int b    = blockIdx.y;
  const int bA   = b * NA;
  const int base = seg * SEGLEN;

  stage_targets(tg, b, tid, traw4);
  const float* traw = (const float*)traw4;
  if (tid < TGT) {
    tb[tid] = make_float4(traw[tid * 5 + 0], traw[tid * 5 + 1],
                          traw[tid * 5 + 2], traw[tid * 5 + 3]);
    tlabel[tid] = traw[tid * 5 + 4];
    aidx[tid]   = NA - 1 - (int)(wsbest[b * TGT + tid] & 0xffffffffull);
  }
  __syncthreads();

  int aend = base + SEGLEN; if (aend > NA) aend = NA;
  for (int a = base + tid; a < aend; a += 256) {
    int   gi = wsix[bA + a];
    float ov = wsov[bA + a];
    // override: forward scan gives "last target wins" on duplicates,
    // matching JAX scatter semantics of gt_index.at[anchor_index].set(...)
#pragma unroll 4
    for (int t = 0; t < TGT; ++t) {
      if (aidx[t] == a) { gi = t; ov = 1.0f; }
    }
    float4 m = tb[gi];
    float4 c = *(const float4*)(an + (size_t)a * 4);   // cx,cy,w,h
    float cx = 0.5f * (m.x + m.z), cy = 0.5f * (m.y + m.w);
    float w  = m.z - m.x,          h  = m.w - m.y;
    float4 o;
    o.x = (cx - c.x) / (0.1f * c.z);
    o.y = (cy - c.y) / (0.1f * c.w);
    o.z = __logf(w / c.z) * 5.0f;    // / VAR1 (0.2)
    o.w = __logf(h / c.w) * 5.0f;
    *(float4*)(loc + (size_t)(bA + a) * 4) = o;
    conf[bA + a] = (ov < THRESH) ? 0 : (int)(tlabel[gi] + 1.0f);
  }
}

extern "C" void kernel_launch(void* const* d_in, const int* in_sizes, int n_in,
                              void* d_out, int out_size, void* d_ws, size_t ws_size,
                              hipStream_t stream) {
  (void)in_sizes; (void)n_in; (void)out_size; (void)ws_size;
  const float* tg = (const float*)d_in[0];   // targets (64,100,5) f32
  const float* an = (const float*)d_in[1];   // anchors (8732,4) f32

  float* loc  = (float*)d_out;                                   // B*A*4 f32
  int*   conf = (int*)((float*)d_out + (size_t)BATCH * NA * 4);  // B*A i32

  char* ws = (char*)d_ws;
  float* wsov = (float*)ws;                                     // B*A f32
  int*   wsix = (int*)(ws + (size_t)BATCH * NA * 4);            // B*A i32
  unsigned long long* wsbest =
      (unsigned long long*)(ws + (size_t)2 * BATCH * NA * 4);   // B*T u64

  init_best_kernel<<<(BATCH * TGT + 255) / 256, 256, 0, stream>>>(wsbest);
  dim3 grid(SEGS, BATCH);
  match_kernel<<<grid, 256, 0, stream>>>(tg, an, wsov, wsix, wsbest);
  encode_kernel<<<grid, 256, 0, stream>>>(tg, an, wsov, wsix, wsbest, loc, conf);
}